// PointTransformerLayer_75213467287641
// MI455X (gfx1250) — compile-verified
//
#include <hip/hip_runtime.h>
#include <math.h>

// Problem dims
#define NPTS   65536
#define CH     256
#define NSAMP  16
#define CSDIM  32
#define RTOT   (NPTS * NSAMP)        // 1048576 rows of (n, j)
#define BN_EPS 1e-5f

typedef float v2f __attribute__((ext_vector_type(2)));
typedef float v8f __attribute__((ext_vector_type(8)));

// ---------------------------------------------------------------------------
// Stats accumulator zeroing
// ---------------------------------------------------------------------------
__global__ void zero_stats_kernel(float* stats, int n) {
    int t = blockIdx.x * blockDim.x + threadIdx.x;
    if (t < n) stats[t] = 0.0f;
}

// ---------------------------------------------------------------------------
// QKV GEMM: out[z] = x @ W[z] + b[z], x:[N,256], W:[256,256]
// block = 256 threads = 8 waves: 4 row-tiles x 2 col-tiles -> 64 rows x 32 cols
// fp32 WMMA 16x16x4; A-tiles staged via double-buffered async-to-LDS B128.
// ---------------------------------------------------------------------------
__global__ __launch_bounds__(256) void qkv_gemm_kernel(
    const float* __restrict__ x,
    const float* __restrict__ wq, const float* __restrict__ bq,
    const float* __restrict__ wk, const float* __restrict__ bk,
    const float* __restrict__ wv, const float* __restrict__ bv,
    float* __restrict__ xq, float* __restrict__ xk, float* __restrict__ xv)
{
    const int z = blockIdx.z;
    const float* __restrict__ W = (z == 0) ? wq : (z == 1) ? wk : wv;
    const float* __restrict__ B = (z == 0) ? bq : (z == 1) ? bk : bv;
    float* __restrict__ O       = (z == 0) ? xq : (z == 1) ? xk : xv;

    // 2 buffers x 64 rows x 32 K; row stride 36 floats = 144B (16B aligned rows)
    __shared__ float s[2][64][36];

    const int tid     = threadIdx.x;
    const int wave    = tid >> 5;
    const int lane    = tid & 31;
    const int laneM   = lane & 15;
    const int laneHi  = lane >> 4;           // 0: K pair {0,1}, 1: K pair {2,3}
    const int rowTile = wave >> 1;           // 0..3
    const int colTile = wave & 1;            // 0..1
    const int row0    = blockIdx.x * 64;
    const int col0    = blockIdx.y * 32 + colTile * 16;

    // per-thread async element: r = row, kk = 4-float group within chunk
    const int ar_r0  = (tid) >> 3;                 // h=0 row
    const int ar_kk0 = ((tid) & 7) * 4;
    const int ar_r1  = (tid + 256) >> 3;           // h=1 row
    const int ar_kk1 = ((tid + 256) & 7) * 4;

    const uint64_t xrow = (uint64_t)(uintptr_t)(x + (size_t)row0 * CH);
    const uint32_t lbase = (uint32_t)(uintptr_t)(&s[0][0][0]);
    const uint32_t bufstride = 64u * 36u * 4u;

    // prologue: stage chunk 0 into buffer 0
    {
        uint32_t d0 = lbase + (uint32_t)((ar_r0 * 36 + ar_kk0) * 4);
        uint32_t o0 = (uint32_t)((ar_r0 * CH + ar_kk0) * 4);
        uint32_t d1 = lbase + (uint32_t)((ar_r1 * 36 + ar_kk1) * 4);
        uint32_t o1 = (uint32_t)((ar_r1 * CH + ar_kk1) * 4);
        asm volatile("global_load_async_to_lds_b128 %0, %1, %2"
                     :: "v"(d0), "v"(o0), "s"(xrow) : "memory");
        asm volatile("global_load_async_to_lds_b128 %0, %1, %2"
                     :: "v"(d1), "v"(o1), "s"(xrow) : "memory");
    }

    v8f acc = {};
    for (int kc = 0; kc < 8; ++kc) {
        const int k0 = kc * 32;
        if (kc + 1 < 8) {
            // stage next chunk into the other buffer
            const uint64_t srcb = xrow + (uint64_t)((k0 + 32) * 4);
            const uint32_t lb   = lbase + (uint32_t)((kc + 1) & 1) * bufstride;
            uint32_t d0 = lb + (uint32_t)((ar_r0 * 36 + ar_kk0) * 4);
            uint32_t o0 = (uint32_t)((ar_r0 * CH + ar_kk0) * 4);
            uint32_t d1 = lb + (uint32_t)((ar_r1 * 36 + ar_kk1) * 4);
            uint32_t o1 = (uint32_t)((ar_r1 * CH + ar_kk1) * 4);
            asm volatile("global_load_async_to_lds_b128 %0, %1, %2"
                         :: "v"(d0), "v"(o0), "s"(srcb) : "memory");
            asm volatile("global_load_async_to_lds_b128 %0, %1, %2"
                         :: "v"(d1), "v"(o1), "s"(srcb) : "memory");
            // own chunk-kc loads complete (in-order) when count drops to 2
            asm volatile("s_wait_asynccnt 0x2" ::: "memory");
        } else {
            asm volatile("s_wait_asynccnt 0x0" ::: "memory");
        }
        __syncthreads();   // all waves' chunk-kc data visible in LDS

        const int buf = kc & 1;
        const int ar  = rowTile * 16 + laneM;
#pragma unroll
        for (int kk = 0; kk < 32; kk += 4) {
            v2f a, b;
            a.x = s[buf][ar][kk + laneHi * 2 + 0];
            a.y = s[buf][ar][kk + laneHi * 2 + 1];
            const int kb = k0 + kk + laneHi * 2;
            b.x = W[(size_t)(kb + 0) * CH + col0 + laneM];
            b.y = W[(size_t)(kb + 1) * CH + col0 + laneM];
            acc = __builtin_amdgcn_wmma_f32_16x16x4_f32(
                false, a, false, b, (short)0, acc, false, false);
        }
        __syncthreads();   // done reading buf before it is overwritten
    }

    const float bias = B[col0 + laneM];
#pragma unroll
    for (int r = 0; r < 8; ++r) {
        int grow = row0 + rowTile * 16 + r + laneHi * 8;
        O[(size_t)grow * CH + col0 + laneM] = acc[r] + bias;
    }
}

// ---------------------------------------------------------------------------
// Stage-1 stats: t1 = (p[idx]-p) @ pw1 + pb1, 3 channels, mean/var over N*NS
// ---------------------------------------------------------------------------
__global__ __launch_bounds__(256) void pr_stats_kernel(
    const float* __restrict__ p, const int* __restrict__ idx,
    const float* __restrict__ pw1, const float* __restrict__ pb1,
    float* __restrict__ stats3)   // [0..2]=sum, [3..5]=sumsq
{
    float s1[3] = {0, 0, 0}, s2[3] = {0, 0, 0};
    const int tid = threadIdx.x;
    for (int r = blockIdx.x * blockDim.x + tid; r < RTOT; r += gridDim.x * blockDim.x) {
        int n = r >> 4;
        int m = idx[r];
        float pr0 = p[m * 3 + 0] - p[n * 3 + 0];
        float pr1 = p[m * 3 + 1] - p[n * 3 + 1];
        float pr2 = p[m * 3 + 2] - p[n * 3 + 2];
#pragma unroll
        for (int e = 0; e < 3; ++e) {
            float t = pr0 * pw1[e] + pr1 * pw1[3 + e] + pr2 * pw1[6 + e] + pb1[e];
            s1[e] += t;
            s2[e] += t * t;
        }
    }
    __shared__ float red[256];
#pragma unroll
    for (int e = 0; e < 3; ++e) {
        red[tid] = s1[e];
        __syncthreads();
        for (int off = 128; off > 0; off >>= 1) {
            if (tid < off) red[tid] += red[tid + off];
            __syncthreads();
        }
        if (tid == 0) atomicAdd(&stats3[e], red[0]);
        __syncthreads();
        red[tid] = s2[e];
        __syncthreads();
        for (int off = 128; off > 0; off >>= 1) {
            if (tid < off) red[tid] += red[tid + off];
            __syncthreads();
        }
        if (tid == 0) atomicAdd(&stats3[3 + e], red[0]);
        __syncthreads();
    }
}

// ---------------------------------------------------------------------------
// Generic BN finalize: scale = g*rsqrt(var+eps); shift = b - mean*scale
// ---------------------------------------------------------------------------
__global__ void finalize_bn_kernel(const float* __restrict__ sums, int nch, float invM,
                                   const float* __restrict__ g, const float* __restrict__ b,
                                   float* __restrict__ scale, float* __restrict__ shift)
{
    int c = blockIdx.x * blockDim.x + threadIdx.x;
    if (c < nch) {
        float mean = sums[c] * invM;
        float var  = sums[nch + c] * invM - mean * mean;
        float rs   = rsqrtf(var + BN_EPS);
        float sc   = rs * g[c];
        scale[c] = sc;
        shift[c] = b[c] - mean * sc;
    }
}

// ---------------------------------------------------------------------------
// r_qk stats: r_qk[c] = xk[idx]-xq+p_feat; sum/sumsq per 256 channels.
// float4 vectorized: thread -> 4 channels x 4 neighbor-streams; LDS reduce.
// ---------------------------------------------------------------------------
__global__ __launch_bounds__(256) void rqk_stats_kernel(
    const float* __restrict__ xq, const float* __restrict__ xk,
    const int* __restrict__ idx, const float* __restrict__ p,
    const float* __restrict__ pw1, const float* __restrict__ pb1,
    const float* __restrict__ scale3, const float* __restrict__ shift3,
    const float* __restrict__ pw2, const float* __restrict__ pb2,
    float* __restrict__ stats1)   // [0..255]=sum, [256..511]=sumsq
{
    __shared__ float rt[16][3];
    __shared__ int   mrow[16];
    __shared__ float red[256][8];

    const int tid = threadIdx.x;
    const int t   = tid & 63;
    const int cg  = t * 4;          // channel base
    const int jg  = tid >> 6;       // neighbor stream 0..3

    const float4 w0  = *(const float4*)(pw2 + cg);
    const float4 w1  = *(const float4*)(pw2 + CH + cg);
    const float4 w2  = *(const float4*)(pw2 + 2 * CH + cg);
    const float4 pbv = *(const float4*)(pb2 + cg);

    float s1x = 0, s1y = 0, s1z = 0, s1w = 0;
    float s2x = 0, s2y = 0, s2z = 0, s2w = 0;

    for (int n = blockIdx.x; n < NPTS; n += gridDim.x) {
        __syncthreads();
        if (tid < 16) {
            int j = tid;
            int m = idx[n * NSAMP + j];
            mrow[j] = m;
            float pr0 = p[m * 3 + 0] - p[n * 3 + 0];
            float pr1 = p[m * 3 + 1] - p[n * 3 + 1];
            float pr2 = p[m * 3 + 2] - p[n * 3 + 2];
#pragma unroll
            for (int e = 0; e < 3; ++e) {
                float tt = pr0 * pw1[e] + pr1 * pw1[3 + e] + pr2 * pw1[6 + e] + pb1[e];
                rt[j][e] = fmaxf(tt * scale3[e] + shift3[e], 0.0f);
            }
        }
        __syncthreads();
        const float4 xqv = *(const float4*)(xq + (size_t)n * CH + cg);
#pragma unroll
        for (int jj = 0; jj < 4; ++jj) {
            int j = jg * 4 + jj;
            const float4 xkv = *(const float4*)(xk + (size_t)mrow[j] * CH + cg);
            float r0 = rt[j][0], r1 = rt[j][1], r2 = rt[j][2];
            float vx = xkv.x - xqv.x + r0 * w0.x + r1 * w1.x + r2 * w2.x + pbv.x;
            float vy = xkv.y - xqv.y + r0 * w0.y + r1 * w1.y + r2 * w2.y + pbv.y;
            float vz = xkv.z - xqv.z + r0 * w0.z + r1 * w1.z + r2 * w2.z + pbv.z;
            float vw = xkv.w - xqv.w + r0 * w0.w + r1 * w1.w + r2 * w2.w + pbv.w;
            s1x += vx; s1y += vy; s1z += vz; s1w += vw;
            s2x += vx * vx; s2y += vy * vy; s2z += vz * vz; s2w += vw * vw;
        }
    }
    red[tid][0] = s1x; red[tid][1] = s1y; red[tid][2] = s1z; red[tid][3] = s1w;
    red[tid][4] = s2x; red[tid][5] = s2y; red[tid][6] = s2z; red[tid][7] = s2w;
    __syncthreads();
    if (tid < 64) {
#pragma unroll
        for (int k = 0; k < 4; ++k) {
            float a = red[t][k] + red[t + 64][k] + red[t + 128][k] + red[t + 192][k];
            atomicAdd(&stats1[cg + k], a);
        }
#pragma unroll
        for (int k = 0; k < 4; ++k) {
            float a = red[t][4 + k] + red[t + 64][4 + k] + red[t + 128][4 + k] + red[t + 192][4 + k];
            atomicAdd(&stats1[CH + cg + k], a);
        }
    }
}

// ---------------------------------------------------------------------------
// w GEMM: w = relu(bn1(r_qk)) @ wc + bc, rows = N*NS, K=256, cols=32.
// r_qk rebuilt on the fly with float4 gathers during LDS staging.
// ---------------------------------------------------------------------------
__global__ __launch_bounds__(256) void w_gemm_kernel(
    const float* __restrict__ xq, const float* __restrict__ xk,
    const int* __restrict__ idx, const float* __restrict__ p,
    const float* __restrict__ pw1, const float* __restrict__ pb1,
    const float* __restrict__ scale3, const float* __restrict__ shift3,
    const float* __restrict__ pw2, const float* __restrict__ pb2,
    const float* __restrict__ scale1, const float* __restrict__ shift1,
    const float* __restrict__ wc, const float* __restrict__ bc,
    float* __restrict__ wout)
{
    __shared__ float s[64][36];      // 144B rows: float4-aligned stores
    __shared__ float rowrt[64][3];
    __shared__ int   rowm[64];
    __shared__ int   rown[64];

    const int tid     = threadIdx.x;
    const int wave    = tid >> 5;
    const int lane    = tid & 31;
    const int laneM   = lane & 15;
    const int laneHi  = lane >> 4;
    const int rowTile = wave >> 1;
    const int colTile = wave & 1;
    const int grow0   = blockIdx.x * 64;
    const int col0    = colTile * 16;

    if (tid < 64) {
        int grow = grow0 + tid;
        int n = grow >> 4;
        int j = grow & 15;
        int m = idx[n * NSAMP + j];
        rowm[tid] = m;
        rown[tid] = n;
        float pr0 = p[m * 3 + 0] - p[n * 3 + 0];
        float pr1 = p[m * 3 + 1] - p[n * 3 + 1];
        float pr2 = p[m * 3 + 2] - p[n * 3 + 2];
#pragma unroll
        for (int e = 0; e < 3; ++e) {
            float t = pr0 * pw1[e] + pr1 * pw1[3 + e] + pr2 * pw1[6 + e] + pb1[e];
            rowrt[tid][e] = fmaxf(t * scale3[e] + shift3[e], 0.0f);
        }
    }

    v8f acc = {};
    for (int k0 = 0; k0 < CH; k0 += 32) {
        __syncthreads();
        // 512 float4 groups (64 rows x 8 groups), 2 per thread
#pragma unroll
        for (int h = 0; h < 2; ++h) {
            int e  = tid + h * 256;
            int r  = e >> 3;
            int kk = (e & 7) * 4;
            int c  = k0 + kk;
            int m  = rowm[r];
            int n  = rown[r];
            const float4 xkv = *(const float4*)(xk + (size_t)m * CH + c);
            const float4 xqv = *(const float4*)(xq + (size_t)n * CH + c);
            const float4 pa  = *(const float4*)(pw2 + c);
            const float4 pbm = *(const float4*)(pw2 + CH + c);
            const float4 pc  = *(const float4*)(pw2 + 2 * CH + c);
            const float4 pb  = *(const float4*)(pb2 + c);
            const float4 sc  = *(const float4*)(scale1 + c);
            const float4 sh  = *(const float4*)(shift1 + c);
            float r0 = rowrt[r][0], r1 = rowrt[r][1], r2 = rowrt[r][2];
            float4 o;
            o.x = fmaxf((xkv.x - xqv.x + r0 * pa.x + r1 * pbm.x + r2 * pc.x + pb.x) * sc.x + sh.x, 0.0f);
            o.y = fmaxf((xkv.y - xqv.y + r0 * pa.y + r1 * pbm.y + r2 * pc.y + pb.y) * sc.y + sh.y, 0.0f);
            o.z = fmaxf((xkv.z - xqv.z + r0 * pa.z + r1 * pbm.z + r2 * pc.z + pb.z) * sc.z + sh.z, 0.0f);
            o.w = fmaxf((xkv.w - xqv.w + r0 * pa.w + r1 * pbm.w + r2 * pc.w + pb.w) * sc.w + sh.w, 0.0f);
            *(float4*)&s[r][kk] = o;
        }
        __syncthreads();
        const int ar = rowTile * 16 + laneM;
#pragma unroll
        for (int kk = 0; kk < 32; kk += 4) {
            v2f a, b;
            a.x = s[ar][kk + laneHi * 2 + 0];
            a.y = s[ar][kk + laneHi * 2 + 1];
            const int kb = k0 + kk + laneHi * 2;
            b.x = wc[(size_t)(kb + 0) * CSDIM + col0 + laneM];
            b.y = wc[(size_t)(kb + 1) * CSDIM + col0 + laneM];
            acc = __builtin_amdgcn_wmma_f32_16x16x4_f32(
                false, a, false, b, (short)0, acc, false, false);
        }
    }
    const float bias = bc[col0 + laneM];
#pragma unroll
    for (int r = 0; r < 8; ++r) {
        int grow = grow0 + rowTile * 16 + r + laneHi * 8;
        wout[(size_t)grow * CSDIM + col0 + laneM] = acc[r] + bias;
    }
}

// ---------------------------------------------------------------------------
// BN2 stats over w: 32 channels, mean/var over N*NS rows.
// ---------------------------------------------------------------------------
__global__ __launch_bounds__(256) void w_stats_kernel(
    const float* __restrict__ w, float* __restrict__ stats2)  // [0..31]=sum,[32..63]=sumsq
{
    const int tid = threadIdx.x;
    const int i   = tid & 31;
    const int g   = tid >> 5;               // 8 row-streams per block
    float s1 = 0.0f, s2 = 0.0f;
    for (int r = blockIdx.x * 8 + g; r < RTOT; r += gridDim.x * 8) {
        float v = w[(size_t)r * CSDIM + i];
        s1 += v;
        s2 += v * v;
    }
    __shared__ float red[256];
    red[tid] = s1;
    __syncthreads();
    if (g == 0) {
        float a = 0.0f;
#pragma unroll
        for (int k = 0; k < 8; ++k) a += red[k * 32 + i];
        atomicAdd(&stats2[i], a);
    }
    __syncthreads();
    red[tid] = s2;
    __syncthreads();
    if (g == 0) {
        float a = 0.0f;
#pragma unroll
        for (int k = 0; k < 8; ++k) a += red[k * 32 + i];
        atomicAdd(&stats2[32 + i], a);
    }
}

// ---------------------------------------------------------------------------
// Final: per point n:  w2 = relu(bn2(w)) @ wd + bd; softmax over neighbors;
// out[n,c] = sum_j softw[j][c&31] * (xv[idx[n,j],c] + p_feat(n,j,c))
// Aggregation float4-vectorized with 4 neighbor-streams + LDS reduce.
// ---------------------------------------------------------------------------
__global__ __launch_bounds__(256) void final_kernel(
    const float* __restrict__ xv, const float* __restrict__ wbuf,
    const int* __restrict__ idx, const float* __restrict__ p,
    const float* __restrict__ pw1, const float* __restrict__ pb1,
    const float* __restrict__ scale3, const float* __restrict__ shift3,
    const float* __restrict__ pw2, const float* __restrict__ pb2,
    const float* __restrict__ scale2, const float* __restrict__ shift2,
    const float* __restrict__ wd, const float* __restrict__ bd,
    float* __restrict__ out)
{
    const int n   = blockIdx.x;
    const int tid = threadIdx.x;

    __shared__ float wrelu[16][33];
    __shared__ float wsoft[16][33];
    __shared__ float rt[16][3];
    __shared__ int   mrow[16];
    __shared__ float red[256][4];

    // Load w rows for this point (float4), apply bn2 + relu
    if (tid < 128) {
        int j = tid >> 3, i = (tid & 7) * 4;
        const float4 v  = *(const float4*)(wbuf + ((size_t)n * NSAMP + j) * CSDIM + i);
        const float4 sc = *(const float4*)(scale2 + i);
        const float4 sh = *(const float4*)(shift2 + i);
        wrelu[j][i + 0] = fmaxf(v.x * sc.x + sh.x, 0.0f);
        wrelu[j][i + 1] = fmaxf(v.y * sc.y + sh.y, 0.0f);
        wrelu[j][i + 2] = fmaxf(v.z * sc.z + sh.z, 0.0f);
        wrelu[j][i + 3] = fmaxf(v.w * sc.w + sh.w, 0.0f);
    }
    if (tid < 16) {
        int j = tid;
        int m = idx[n * NSAMP + j];
        mrow[j] = m;
        float pr0 = p[m * 3 + 0] - p[n * 3 + 0];
        float pr1 = p[m * 3 + 1] - p[n * 3 + 1];
        float pr2 = p[m * 3 + 2] - p[n * 3 + 2];
#pragma unroll
        for (int e = 0; e < 3; ++e) {
            float t = pr0 * pw1[e] + pr1 * pw1[3 + e] + pr2 * pw1[6 + e] + pb1[e];
            rt[j][e] = fmaxf(t * scale3[e] + shift3[e], 0.0f);
        }
    }
    __syncthreads();

    // w2 = wrelu @ wd + bd  (16x32 @ 32x32); each thread computes 2 outputs
#pragma unroll
    for (int h = 0; h < 2; ++h) {
        int e = tid + h * 256;
        int j = e >> 5, i = e & 31;
        float a = bd[i];
#pragma unroll
        for (int k = 0; k < CSDIM; ++k) a += wrelu[j][k] * wd[k * CSDIM + i];
        wsoft[j][i] = a;
    }
    __syncthreads();

    // softmax over neighbor dim per channel i
    if (tid < 32) {
        const int i = tid;
        float mx = -3.4e38f;
#pragma unroll
        for (int j = 0; j < NSAMP; ++j) mx = fmaxf(mx, wsoft[j][i]);
        float sum = 0.0f;
#pragma unroll
        for (int j = 0; j < NSAMP; ++j) {
            float ev = __expf(wsoft[j][i] - mx);
            wsoft[j][i] = ev;
            sum += ev;
        }
        float inv = 1.0f / sum;
#pragma unroll
        for (int j = 0; j < NSAMP; ++j) wsoft[j][i] *= inv;
    }
    __syncthreads();

    // aggregate: thread -> 4 channels (c0..c0+3), 4 neighbor-streams
    const int t  = tid & 63;
    const int q  = tid >> 6;
    const int c0 = t * 4;
    const int i0 = c0 & 31;
    const float4 w0  = *(const float4*)(pw2 + c0);
    const float4 w1  = *(const float4*)(pw2 + CH + c0);
    const float4 w2  = *(const float4*)(pw2 + 2 * CH + c0);
    const float4 pbv = *(const float4*)(pb2 + c0);
    float a0 = 0, a1 = 0, a2 = 0, a3 = 0;
#pragma unroll
    for (int jj = 0; jj < 4; ++jj) {
        int j = q * 4 + jj;
        const float4 xvv = *(const float4*)(xv + (size_t)mrow[j] * CH + c0);
        float r0 = rt[j][0], r1 = rt[j][1], r2 = rt[j][2];
        float sw0 = wsoft[j][i0 + 0], sw1 = wsoft[j][i0 + 1];
        float sw2 = wsoft[j][i0 + 2], sw3 = wsoft[j][i0 + 3];
        a0 += sw0 * (xvv.x + r0 * w0.x + r1 * w1.x + r2 * w2.x + pbv.x);
        a1 += sw1 * (xvv.y + r0 * w0.y + r1 * w1.y + r2 * w2.y + pbv.y);
        a2 += sw2 * (xvv.z + r0 * w0.z + r1 * w1.z + r2 * w2.z + pbv.z);
        a3 += sw3 * (xvv.w + r0 * w0.w + r1 * w1.w + r2 * w2.w + pbv.w);
    }
    red[tid][0] = a0; red[tid][1] = a1; red[tid][2] = a2; red[tid][3] = a3;
    __syncthreads();
    if (tid < 64) {
        float4 o;
        o.x = red[t][0] + red[t + 64][0] + red[t + 128][0] + red[t + 192][0];
        o.y = red[t][1] + red[t + 64][1] + red[t + 128][1] + red[t + 192][1];
        o.z = red[t][2] + red[t + 64][2] + red[t + 128][2] + red[t + 192][2];
        o.w = red[t][3] + red[t + 64][3] + red[t + 128][3] + red[t + 192][3];
        *(float4*)(out + (size_t)n * CH + c0) = o;
    }
}

// ---------------------------------------------------------------------------
// Host launcher
// ---------------------------------------------------------------------------
extern "C" void kernel_launch(void* const* d_in, const int* in_sizes, int n_in,
                              void* d_out, int out_size, void* d_ws, size_t ws_size,
                              hipStream_t stream) {
    const float* p      = (const float*)d_in[0];
    const float* x      = (const float*)d_in[1];
    const int*   idx    = (const int*)  d_in[2];
    const float* wq     = (const float*)d_in[3];
    const float* bq     = (const float*)d_in[4];
    const float* wk     = (const float*)d_in[5];
    const float* bk     = (const float*)d_in[6];
    const float* wv     = (const float*)d_in[7];
    const float* bv     = (const float*)d_in[8];
    const float* pw1    = (const float*)d_in[9];
    const float* pb1    = (const float*)d_in[10];
    const float* pg1    = (const float*)d_in[11];
    const float* pbeta1 = (const float*)d_in[12];
    const float* pw2    = (const float*)d_in[13];
    const float* pb2    = (const float*)d_in[14];
    const float* wg1    = (const float*)d_in[15];
    const float* wbeta1 = (const float*)d_in[16];
    const float* wc     = (const float*)d_in[17];
    const float* bc     = (const float*)d_in[18];
    const float* wg2    = (const float*)d_in[19];
    const float* wbeta2 = (const float*)d_in[20];
    const float* wd     = (const float*)d_in[21];
    const float* bd     = (const float*)d_in[22];
    float* out = (float*)d_out;

    // Workspace layout (floats)
    float* ws = (float*)d_ws;
    size_t o = 0;
    float* xq   = ws + o; o += (size_t)NPTS * CH;       // 16M
    float* xk   = ws + o; o += (size_t)NPTS * CH;       // 16M
    float* xv   = ws + o; o += (size_t)NPTS * CH;       // 16M
    float* wbuf = ws + o; o += (size_t)RTOT * CSDIM;    // 32M
    float* stats3 = ws + o; o += 8;                     // sum[3], sumsq[3]
    float* stats1 = ws + o; o += 2 * CH;                // sum[256], sumsq[256]
    float* stats2 = ws + o; o += 2 * CSDIM;             // sum[32], sumsq[32]
    float* scale3 = ws + o; o += 4;
    float* shift3 = ws + o; o += 4;
    float* scale1 = ws + o; o += CH;
    float* shift1 = ws + o; o += CH;
    float* scale2 = ws + o; o += CSDIM;
    float* shift2 = ws + o; o += CSDIM;

    const float invM = 1.0f / (float)RTOT;
    const int nstats = 8 + 2 * CH + 2 * CSDIM;

    // 1) zero stat accumulators
    zero_stats_kernel<<<(nstats + 255) / 256, 256, 0, stream>>>(stats3, nstats);

    // 2) QKV GEMMs (fp32 WMMA + async-to-LDS double buffering)
    qkv_gemm_kernel<<<dim3(NPTS / 64, CH / 32, 3), 256, 0, stream>>>(
        x, wq, bq, wk, bk, wv, bv, xq, xk, xv);

    // 3) p_r stage-1 stats + finalize (3 ch)
    pr_stats_kernel<<<1024, 256, 0, stream>>>(p, idx, pw1, pb1, stats3);
    finalize_bn_kernel<<<1, 256, 0, stream>>>(stats3, 3, invM, pg1, pbeta1, scale3, shift3);

    // 4) r_qk stats + finalize (256 ch)
    rqk_stats_kernel<<<2048, 256, 0, stream>>>(
        xq, xk, idx, p, pw1, pb1, scale3, shift3, pw2, pb2, stats1);
    finalize_bn_kernel<<<1, 256, 0, stream>>>(stats1, CH, invM, wg1, wbeta1, scale1, shift1);

    // 5) w = relu(bn1(r_qk)) @ wc + bc (fp32 WMMA, r_qk fused on the fly)
    w_gemm_kernel<<<RTOT / 64, 256, 0, stream>>>(
        xq, xk, idx, p, pw1, pb1, scale3, shift3, pw2, pb2,
        scale1, shift1, wc, bc, wbuf);

    // 6) w stats + finalize (32 ch)
    w_stats_kernel<<<1024, 256, 0, stream>>>(wbuf, stats2);
    finalize_bn_kernel<<<1, 256, 0, stream>>>(stats2, CSDIM, invM, wg2, wbeta2, scale2, shift2);

    // 7) final: bn2+relu -> 32x32 matmul -> softmax -> gather-aggregate
    final_kernel<<<NPTS, 256, 0, stream>>>(
        xv, wbuf, idx, p, pw1, pb1, scale3, shift3, pw2, pb2,
        scale2, shift2, wd, bd, out);

    (void)in_sizes; (void)n_in; (void)out_size; (void)ws_size;
}